// GNNComplete_29643864277578
// MI455X (gfx1250) — compile-verified
//
#include <hip/hip_runtime.h>
#include <hip/hip_bf16.h>

#define DD 128
#define LDS_PITCH 132   // 132 mod 64banks = 4 -> conflict-free A-fragment reads

typedef __attribute__((ext_vector_type(2))) float v2f;
typedef __attribute__((ext_vector_type(8))) float v8f;

// ---------------------------------------------------------------------------
// hx[N,128] = h[N,128] @ W[128,128] + b   via V_WMMA_F32_16X16X4_F32
// block = 256 thr = 8 waves; block -> 16 rows, wave w -> cols [16w,16w+16)
// Epilogue stages the 16x128 result tile through LDS for coalesced stores.
// ---------------------------------------------------------------------------
__global__ __launch_bounds__(256) void k_gemm_hx(
    const float* __restrict__ h, const float* __restrict__ Wl,
    const float* __restrict__ bl, float* __restrict__ hx, int N)
{
    __shared__ __align__(16) float sh[16 * LDS_PITCH];
    const int tid = threadIdx.x;
    const int r0  = blockIdx.x * 16;

    // cooperative 16x128 tile load (float4), zero-fill OOB rows
    for (int i = tid; i < 16 * 32; i += 256) {
        const int r  = i >> 5;
        const int c4 = i & 31;
        float4 v = make_float4(0.f, 0.f, 0.f, 0.f);
        if (r0 + r < N) v = ((const float4*)(h + (size_t)(r0 + r) * DD))[c4];
        *(float4*)&sh[r * LDS_PITCH + c4 * 4] = v;   // 528B row pitch, 16B aligned
    }
    __syncthreads();

    const int wv    = tid >> 5;       // 0..7 : 16-col tile
    const int lane  = tid & 31;
    const int m     = lane & 15;      // A row
    const int khalf = lane >> 4;      // 0 -> K{0,1}, 1 -> K{2,3}
    const int n     = wv * 16 + (lane & 15);  // output column

    v8f acc = {};
    #pragma unroll 8
    for (int ks = 0; ks < 32; ++ks) {
        const int kk = ks * 4 + khalf * 2;
        v2f a, bb;
        a.x  = sh[m * LDS_PITCH + kk];
        a.y  = sh[m * LDS_PITCH + kk + 1];
        bb.x = Wl[(size_t)kk * DD + n];
        bb.y = Wl[(size_t)(kk + 1) * DD + n];
        acc = __builtin_amdgcn_wmma_f32_16x16x4_f32(
                  false, a, false, bb, (short)0, acc, false, false);
    }

    // stage D tile back through LDS: lane's VGPR j holds (M = j + 8*khalf, N = n)
    const float bias = bl[n];
    __syncthreads();   // all waves done reading A fragments from sh
    #pragma unroll
    for (int j = 0; j < 8; ++j)
        sh[(khalf * 8 + j) * LDS_PITCH + n] = acc[j] + bias;
    __syncthreads();

    // coalesced float4 store with single row guard
    for (int i = tid; i < 16 * 32; i += 256) {
        const int r  = i >> 5;
        const int c4 = i & 31;
        if (r0 + r < N)
            ((float4*)(hx + (size_t)(r0 + r) * DD))[c4] =
                *(const float4*)&sh[r * LDS_PITCH + c4 * 4];
    }
}

// ---------------------------------------------------------------------------
// degree / norm precompute
// ---------------------------------------------------------------------------
__global__ void k_degree(const int* __restrict__ row, float* __restrict__ deg, int E)
{
    const int e = blockIdx.x * blockDim.x + threadIdx.x;
    if (e < E) unsafeAtomicAdd(&deg[row[e]], 1.0f);
}

__global__ void k_dinv(float* __restrict__ deg, float* __restrict__ dinv, int N)
{
    const int i = blockIdx.x * blockDim.x + threadIdx.x;
    if (i < N) {
        const float d = deg[i] + 1.0f;   // +1 self
        deg[i]  = d;
        dinv[i] = rsqrtf(d);
    }
}

// ---------------------------------------------------------------------------
// edge message + scatter-add:  agg[col] += norm * relu(hx[row] + bond[eattr])
// one wave per edge, lane handles 4 channels (float4) -> 512B coalesced/edge
// ---------------------------------------------------------------------------
__global__ __launch_bounds__(256) void k_scatter(
    const float* __restrict__ hx, const float* __restrict__ dinv,
    const int* __restrict__ row, const int* __restrict__ col,
    const int* __restrict__ eattr, const float* __restrict__ bondl,
    float* __restrict__ agg, int E)
{
    const int gw   = (blockIdx.x * 256 + threadIdx.x) >> 5;
    const int lane = threadIdx.x & 31;
    if (gw >= E) return;
    const int r = row[gw];
    const int c = col[gw];
    const float nrm = dinv[r] * dinv[c];
    const float4 hv = ((const float4*)(hx + (size_t)r * DD))[lane];
    const float4 ev = ((const float4*)(bondl + (size_t)eattr[gw] * DD))[lane];
    float* ag = agg + (size_t)c * DD + lane * 4;
    unsafeAtomicAdd(ag + 0, nrm * fmaxf(hv.x + ev.x, 0.f));
    unsafeAtomicAdd(ag + 1, nrm * fmaxf(hv.y + ev.y, 0.f));
    unsafeAtomicAdd(ag + 2, nrm * fmaxf(hv.z + ev.z, 0.f));
    unsafeAtomicAdd(ag + 3, nrm * fmaxf(hv.w + ev.w, 0.f));
}

// ---------------------------------------------------------------------------
// h2 = agg + relu(hx + root)/deg  (in place in agg) + BN sum/sumsq accumulate
// channel = tid & 127 (grid stride is a multiple of 128)
// ---------------------------------------------------------------------------
__global__ __launch_bounds__(256) void k_self_bn(
    const float* __restrict__ hx, float* __restrict__ h2,
    const float* __restrict__ rootl, const float* __restrict__ deg,
    float* __restrict__ bnsum, float* __restrict__ bnsumsq, int N)
{
    __shared__ float s1[256];
    __shared__ float s2[256];
    const int tid = threadIdx.x;
    const int c   = tid & 127;
    const float rc = rootl[c];
    float lsum = 0.f, lsq = 0.f;
    const size_t total  = (size_t)N * DD;
    const size_t stride = (size_t)gridDim.x * 256;
    for (size_t i = (size_t)blockIdx.x * 256 + tid; i < total; i += stride) {
        const size_t node = i >> 7;
        const float v = h2[i] + fmaxf(hx[i] + rc, 0.f) / deg[node];
        h2[i] = v;
        lsum += v;
        lsq  += v * v;
    }
    s1[tid] = lsum;
    s2[tid] = lsq;
    __syncthreads();
    if (tid < 128) {
        unsafeAtomicAdd(&bnsum[c],   s1[tid] + s1[tid + 128]);
        unsafeAtomicAdd(&bnsumsq[c], s2[tid] + s2[tid + 128]);
    }
}

// ---------------------------------------------------------------------------
// h = relu( (h2 - mu) * rsqrt(var+eps) * gamma + beta )
// ---------------------------------------------------------------------------
__global__ __launch_bounds__(256) void k_bn_apply(
    float* __restrict__ h2, const float* __restrict__ bnsum,
    const float* __restrict__ bnsumsq, const float* __restrict__ gammal,
    const float* __restrict__ betal, int N)
{
    const size_t i = (size_t)blockIdx.x * 256 + threadIdx.x;
    if (i >= (size_t)N * DD) return;
    const int c = (int)(i & 127);
    const float invN = 1.0f / (float)N;
    const float mu  = bnsum[c] * invN;
    const float var = bnsumsq[c] * invN - mu * mu;
    const float v = (h2[i] - mu) * rsqrtf(var + 1e-5f) * gammal[c] + betal[c];
    h2[i] = fmaxf(v, 0.f);
}

// ---------------------------------------------------------------------------
// global mean pool: pool[g] += h[node]; cnt[g] += 1
// ---------------------------------------------------------------------------
__global__ __launch_bounds__(256) void k_pool(
    const float* __restrict__ h, const int* __restrict__ batch,
    float* __restrict__ pool, float* __restrict__ cnt, int N)
{
    const int node = (blockIdx.x * 256 + threadIdx.x) >> 5;
    const int lane = threadIdx.x & 31;
    if (node >= N) return;
    const int g = batch[node];
    const float4 v = ((const float4*)(h + (size_t)node * DD))[lane];
    float* p = pool + (size_t)g * DD + lane * 4;
    unsafeAtomicAdd(p + 0, v.x);
    unsafeAtomicAdd(p + 1, v.y);
    unsafeAtomicAdd(p + 2, v.z);
    unsafeAtomicAdd(p + 3, v.w);
    if (lane == 0) unsafeAtomicAdd(&cnt[g], 1.0f);
}

// ---------------------------------------------------------------------------
// out[G,T] = (pool/cnt) @ W_out + b_out   (tiny: G=128, T=10)
// ---------------------------------------------------------------------------
__global__ __launch_bounds__(32) void k_out(
    const float* __restrict__ pool, const float* __restrict__ cnt,
    const float* __restrict__ Wout, const float* __restrict__ bout,
    float* __restrict__ out, int T)
{
    const int g = blockIdx.x;
    const int t = threadIdx.x;
    if (t >= T) return;
    const float inv = 1.0f / fmaxf(cnt[g], 1.0f);
    float acc = bout[t];
    #pragma unroll 4
    for (int d = 0; d < DD; ++d)
        acc += pool[(size_t)g * DD + d] * inv * Wout[d * T + t];
    out[g * T + t] = acc;
}

// ---------------------------------------------------------------------------
extern "C" void kernel_launch(void* const* d_in, const int* in_sizes, int n_in,
                              void* d_out, int out_size, void* d_ws, size_t ws_size,
                              hipStream_t stream)
{
    const float* x     = (const float*)d_in[0];
    const int*   eidx  = (const int*)d_in[1];
    const int*   eattr = (const int*)d_in[2];
    const int*   batch = (const int*)d_in[3];
    const float* W     = (const float*)d_in[4];
    const float* b     = (const float*)d_in[5];
    const float* root  = (const float*)d_in[6];
    const float* bond  = (const float*)d_in[7];
    const float* gamma = (const float*)d_in[8];
    const float* beta  = (const float*)d_in[9];
    const float* Wout  = (const float*)d_in[10];
    const float* bout  = (const float*)d_in[11];
    float*       out   = (float*)d_out;

    const int E  = in_sizes[2];
    const int N  = in_sizes[3];
    const int T  = in_sizes[11];
    const int L  = in_sizes[5] / DD;
    const int NB = in_sizes[7] / (L * DD);
    const int G  = out_size / T;

    const int* row = eidx;
    const int* col = eidx + E;

    // workspace layout (floats)
    float* ws      = (float*)d_ws;
    float* bufA    = ws;                         // h   / agg  [N*D]
    float* bufB    = bufA + (size_t)N * DD;      // hx          [N*D]
    float* deg     = bufB + (size_t)N * DD;      // [N]
    float* dinv    = deg + N;                    // [N]
    float* bnsum   = dinv + N;                   // [D]
    float* bnsumsq = bnsum + DD;                 // [D]
    float* pool    = bnsumsq + DD;               // [G*D]
    float* cnt     = pool + (size_t)G * DD;      // [G]

    // h = x ; deg precompute
    hipMemcpyAsync(bufA, x, (size_t)N * DD * sizeof(float),
                   hipMemcpyDeviceToDevice, stream);
    hipMemsetAsync(deg, 0, (size_t)N * sizeof(float), stream);
    k_degree<<<(E + 255) / 256, 256, 0, stream>>>(row, deg, E);
    k_dinv<<<(N + 255) / 256, 256, 0, stream>>>(deg, dinv, N);

    const int gemm_blocks = (N + 15) / 16;
    const int scat_blocks = (E * 32 + 255) / 256;
    const int elem_blocks = (int)(((size_t)N * DD + 255) / 256);

    for (int l = 0; l < L; ++l) {
        // hx = h @ W[l] + b[l]       (WMMA f32)
        k_gemm_hx<<<gemm_blocks, 256, 0, stream>>>(
            bufA, W + (size_t)l * DD * DD, b + (size_t)l * DD, bufB, N);
        // agg = 0
        hipMemsetAsync(bufA, 0, (size_t)N * DD * sizeof(float), stream);
        // agg[col] += norm * relu(hx[row] + bond)
        k_scatter<<<scat_blocks, 256, 0, stream>>>(
            bufB, dinv, row, col, eattr, bond + (size_t)l * NB * DD, bufA, E);
        // self/root term + BN stats
        hipMemsetAsync(bnsum, 0, 2 * DD * sizeof(float), stream);
        k_self_bn<<<1024, 256, 0, stream>>>(
            bufB, bufA, root + (size_t)l * DD, deg, bnsum, bnsumsq, N);
        // BN normalize + relu
        k_bn_apply<<<elem_blocks, 256, 0, stream>>>(
            bufA, bnsum, bnsumsq, gamma + (size_t)l * DD, beta + (size_t)l * DD, N);
    }

    // mean pool + output head
    hipMemsetAsync(pool, 0, ((size_t)G * DD + G) * sizeof(float), stream);
    k_pool<<<(N * 32 + 255) / 256, 256, 0, stream>>>(bufA, batch, pool, cnt, N);
    k_out<<<G, 32, 0, stream>>>(pool, cnt, Wout, bout, out, T);
}